// GATLayer_39694087750290
// MI455X (gfx1250) — compile-verified
//
#include <hip/hip_runtime.h>
#include <climits>

typedef __attribute__((ext_vector_type(2))) float v2f;
typedef __attribute__((ext_vector_type(8))) float v8f;

#define HC 256          // H*C output channels
#define KD 256          // IN
#define ATT_SLOPE 0.2f
#define POST_SLOPE 0.01f

// Order-preserving float<->int map so atomicMax(int) == float max (no NaNs in data).
__device__ __forceinline__ int f2ord(float f) {
    int b = __float_as_int(f);
    return b >= 0 ? b : (b ^ 0x7fffffff);
}
__device__ __forceinline__ float ord2f(int s) {
    return __int_as_float(s >= 0 ? s : (s ^ 0x7fffffff));
}

// ---------------- init: zero accum/denom, mmax = -inf (ordered-int INT_MIN) -------------
__global__ __launch_bounds__(256) void gat_init(float* __restrict__ accum,
                                                float* __restrict__ denom,
                                                int* __restrict__ mmax, int n) {
    int idx = blockIdx.x * 256 + threadIdx.x;
    if (idx < n * HC) accum[idx] = 0.0f;
    if (idx < n * 8) { denom[idx] = 0.0f; mmax[idx] = INT_MIN; }
}

// ---------------- fused triple GEMM  Y = X @ W^T  (f32 WMMA 16x16x4) -------------------
// grid.x = ceil(M/64), grid.y = 3 selects (Wl,Wr,Wskip) -> (xl,xr,skip)
//
// LDS tiles are stored in *fragment order*: each lane's operand stream for one
// 16x16 tile over a K=32 chunk is 16 contiguous f32 at lane stride 20 dwords
// (16B-aligned for ds_load_b128; 20*l mod 64 spreads start banks).
//
// V_WMMA_F32_16X16X4_F32 operand layout (ISA 7.12.2, 32-bit A 16x4):
//   A: lane = (m&15) + 16*((k>>1)&1), vgpr = k&1
//   B: lane = (n&15) + 16*((k>>1)&1), vgpr = k&1   (row-striped analogue)
//   D: M = vgpr + 8*(lane>=16), N = lane&15
#define LSTRIDE 20              // dwords per lane within a tile
#define TILEF  (32 * LSTRIDE)   // floats per 16x16-tile chunk buffer

__device__ __forceinline__ int frag_off(int rc, int k) {
    // rc = row (A) or col (B) within 16-tile is rc&15; k in [0,32)
    return ((rc & 15) + 16 * ((k >> 1) & 1)) * LSTRIDE + ((k >> 2) << 1) + (k & 1);
}

__global__ __launch_bounds__(256) void gat_gemm3(
    const float* __restrict__ X,
    const float* __restrict__ W0, const float* __restrict__ W1, const float* __restrict__ W2,
    float* __restrict__ Y0, float* __restrict__ Y1, float* __restrict__ Y2,
    int M)
{
    const float* W = (blockIdx.y == 0) ? W0 : (blockIdx.y == 1 ? W1 : W2);
    float*       Y = (blockIdx.y == 0) ? Y0 : (blockIdx.y == 1 ? Y1 : Y2);

    __shared__ float XsF[4 * TILEF];    // 4 M-tiles  x (32 lanes x 20)
    __shared__ float WsF[16 * TILEF];   // 16 N-tiles x (32 lanes x 20)

    const int t    = threadIdx.x;
    const int lane = t & 31;
    const int wv   = t >> 5;
    const int m0   = blockIdx.x * 64;
    const int nb16 = wv * 2;            // this wave's two 16-column tiles

    v8f acc[4][2];
    #pragma unroll
    for (int a = 0; a < 4; ++a)
        #pragma unroll
        for (int b = 0; b < 2; ++b)
            acc[a][b] = (v8f)0.0f;

    for (int k0 = 0; k0 < KD; k0 += 32) {
        // --- stage X tile (64 rows x 32 K), 8 contiguous elems per thread ---
        {
            int base = t * 8;
            int r = base >> 5;
            int c = base & 31;
            int row = m0 + r;
            float4 f0 = make_float4(0.f, 0.f, 0.f, 0.f), f1 = f0;
            if (row < M) {
                const float* src = X + (size_t)row * KD + k0 + c;
                f0 = ((const float4*)src)[0];
                f1 = ((const float4*)src)[1];
                if (k0 + 32 < KD) __builtin_prefetch(src + 32, 0, 0);
            }
            float* xt = XsF + (r >> 4) * TILEF;
            xt[frag_off(r, c + 0)] = f0.x; xt[frag_off(r, c + 1)] = f0.y;
            xt[frag_off(r, c + 2)] = f0.z; xt[frag_off(r, c + 3)] = f0.w;
            xt[frag_off(r, c + 4)] = f1.x; xt[frag_off(r, c + 5)] = f1.y;
            xt[frag_off(r, c + 6)] = f1.z; xt[frag_off(r, c + 7)] = f1.w;
        }
        // --- stage W tile (256 N-rows x 32 K): thread t loads row t ---
        {
            const float* src = W + (size_t)t * KD + k0;
            float* wt = WsF + (t >> 4) * TILEF;
            if (k0 + 32 < KD) __builtin_prefetch(src + 32, 0, 0);
            #pragma unroll
            for (int q = 0; q < 8; ++q) {
                float4 f = ((const float4*)src)[q];
                int c = q * 4;
                wt[frag_off(t, c + 0)] = f.x; wt[frag_off(t, c + 1)] = f.y;
                wt[frag_off(t, c + 2)] = f.z; wt[frag_off(t, c + 3)] = f.w;
            }
        }
        __syncthreads();

        // --- B operand streams: 2 N-tiles x 16 f32, contiguous per lane ---
        float bx[2][16];
        #pragma unroll
        for (int nt = 0; nt < 2; ++nt) {
            const float4* bp = (const float4*)(WsF + (nb16 + nt) * TILEF + lane * LSTRIDE);
            #pragma unroll
            for (int q = 0; q < 4; ++q) {
                float4 f = bp[q];
                bx[nt][q * 4 + 0] = f.x; bx[nt][q * 4 + 1] = f.y;
                bx[nt][q * 4 + 2] = f.z; bx[nt][q * 4 + 3] = f.w;
            }
        }
        // --- A streams + 8 K-steps of f32 WMMA per M-tile ---
        #pragma unroll
        for (int mt = 0; mt < 4; ++mt) {
            float ax[16];
            const float4* apf = (const float4*)(XsF + mt * TILEF + lane * LSTRIDE);
            #pragma unroll
            for (int q = 0; q < 4; ++q) {
                float4 f = apf[q];
                ax[q * 4 + 0] = f.x; ax[q * 4 + 1] = f.y;
                ax[q * 4 + 2] = f.z; ax[q * 4 + 3] = f.w;
            }
            #pragma unroll
            for (int s = 0; s < 8; ++s) {
                v2f av; av[0] = ax[s * 2]; av[1] = ax[s * 2 + 1];
                v2f b0; b0[0] = bx[0][s * 2]; b0[1] = bx[0][s * 2 + 1];
                v2f b1; b1[0] = bx[1][s * 2]; b1[1] = bx[1][s * 2 + 1];
                acc[mt][0] = __builtin_amdgcn_wmma_f32_16x16x4_f32(
                    false, av, false, b0, (short)0, acc[mt][0], false, false);
                acc[mt][1] = __builtin_amdgcn_wmma_f32_16x16x4_f32(
                    false, av, false, b1, (short)0, acc[mt][1], false, false);
            }
        }
        __syncthreads();
    }

    // --- store D: M = r + 8*(lane>=16), N = lane&15 ---
    const int colD = lane & 15;
    const int rofs = (lane & 16) ? 8 : 0;
    #pragma unroll
    for (int mt = 0; mt < 4; ++mt)
        #pragma unroll
        for (int nt = 0; nt < 2; ++nt)
            #pragma unroll
            for (int r = 0; r < 8; ++r) {
                int row = m0 + mt * 16 + r + rofs;
                if (row < M)
                    Y[(size_t)row * HC + (nb16 + nt) * 16 + colD] = acc[mt][nt][r];
            }
}

// ------------- edge logits: a^T leaky(xl[j]+xr[i]) ; deterministic segment max ---------
__global__ __launch_bounds__(256) void gat_logits(
    const float* __restrict__ xl, const float* __restrict__ xr,
    const float* __restrict__ att, const int* __restrict__ ei,
    int E, int n, float* __restrict__ logits, int* __restrict__ mmax)
{
    __shared__ float satt[HC];
    satt[threadIdx.x] = att[threadIdx.x];
    __syncthreads();

    const int lane = threadIdx.x & 31;
    const int wv   = threadIdx.x >> 5;
    const int ET   = E + n;
    const int stride = gridDim.x * 8;

    for (int e = blockIdx.x * 8 + wv; e < ET; e += stride) {
        int j, i;
        if (e < E) { j = ei[e]; i = ei[E + e]; } else { j = i = e - E; }
        const float4* pj = (const float4*)(xl + (size_t)j * HC) + lane * 2;
        const float4* pi = (const float4*)(xr + (size_t)i * HC) + lane * 2;
        float4 a0 = pj[0], a1 = pj[1];
        float4 b0 = pi[0], b1 = pi[1];
        const float* ap = satt + lane * 8;
        float s = 0.0f, z;
        z = a0.x + b0.x; s += ap[0] * (z > 0.f ? z : ATT_SLOPE * z);
        z = a0.y + b0.y; s += ap[1] * (z > 0.f ? z : ATT_SLOPE * z);
        z = a0.z + b0.z; s += ap[2] * (z > 0.f ? z : ATT_SLOPE * z);
        z = a0.w + b0.w; s += ap[3] * (z > 0.f ? z : ATT_SLOPE * z);
        z = a1.x + b1.x; s += ap[4] * (z > 0.f ? z : ATT_SLOPE * z);
        z = a1.y + b1.y; s += ap[5] * (z > 0.f ? z : ATT_SLOPE * z);
        z = a1.z + b1.z; s += ap[6] * (z > 0.f ? z : ATT_SLOPE * z);
        z = a1.w + b1.w; s += ap[7] * (z > 0.f ? z : ATT_SLOPE * z);
        // lanes 4m..4m+3 cover head m: reduce within groups of 4
        s += __shfl_xor(s, 1, 32);
        s += __shfl_xor(s, 2, 32);
        if ((lane & 3) == 0) {
            int h = lane >> 2;
            logits[(size_t)e * 8 + h] = s;
            atomicMax(&mmax[i * 8 + h], f2ord(s));
        }
    }
}

// ------------- p = exp(logit - max); accumulate denom and unnormalized message ---------
__global__ __launch_bounds__(256) void gat_accum(
    const float* __restrict__ xl, const float* __restrict__ logits,
    const int* __restrict__ mmax, const int* __restrict__ ei,
    int E, int n, float* __restrict__ denom, float* __restrict__ accum)
{
    const int lane = threadIdx.x & 31;
    const int wv   = threadIdx.x >> 5;
    const int ET   = E + n;
    const int stride = gridDim.x * 8;
    const int h    = lane >> 2;

    for (int e = blockIdx.x * 8 + wv; e < ET; e += stride) {
        int j, i;
        if (e < E) { j = ei[e]; i = ei[E + e]; } else { j = i = e - E; }
        float lg = logits[(size_t)e * 8 + h];
        float mv = ord2f(mmax[i * 8 + h]);
        float p  = __expf(lg - mv);
        if ((lane & 3) == 0) atomicAdd(&denom[i * 8 + h], p);
        const float4* pj = (const float4*)(xl + (size_t)j * HC) + lane * 2;
        float4 a0 = pj[0], a1 = pj[1];
        float* ac = accum + (size_t)i * HC + lane * 8;
        atomicAdd(ac + 0, p * a0.x); atomicAdd(ac + 1, p * a0.y);
        atomicAdd(ac + 2, p * a0.z); atomicAdd(ac + 3, p * a0.w);
        atomicAdd(ac + 4, p * a1.x); atomicAdd(ac + 5, p * a1.y);
        atomicAdd(ac + 6, p * a1.z); atomicAdd(ac + 7, p * a1.w);
    }
}

// ------------- finalize: normalize, +bias, leaky(0.01), +skip --------------------------
__global__ __launch_bounds__(256) void gat_finalize(
    const float* __restrict__ accum, const float* __restrict__ denom,
    const float* __restrict__ skip, const float* __restrict__ bias,
    float* __restrict__ out, int n)
{
    int idx = blockIdx.x * 256 + threadIdx.x;
    if (idx >= n * HC) return;
    int c = idx & (HC - 1);
    int i = idx >> 8;
    int hh = c >> 5;
    float v = accum[idx] / denom[i * 8 + hh] + bias[c];
    v = (v > 0.f) ? v : POST_SLOPE * v;
    out[idx] = v + skip[idx];
}

extern "C" void kernel_launch(void* const* d_in, const int* in_sizes, int n_in,
                              void* d_out, int out_size, void* d_ws, size_t ws_size,
                              hipStream_t stream)
{
    const float* x     = (const float*)d_in[0];
    const float* Wl    = (const float*)d_in[1];
    const float* Wr    = (const float*)d_in[2];
    const float* att   = (const float*)d_in[3];
    const float* bias  = (const float*)d_in[4];
    const float* Wskip = (const float*)d_in[5];
    const int*   ei    = (const int*)d_in[6];

    const int n  = in_sizes[0] / KD;   // 50000
    const int E  = in_sizes[6] / 2;    // 800000
    const int ET = E + n;

    float* ws = (float*)d_ws;
    size_t nHC = (size_t)n * HC;
    float* xl     = ws;
    float* xr     = xl + nHC;
    float* skip   = xr + nHC;
    float* accum  = skip + nHC;
    float* logits = accum + nHC;
    float* denom  = logits + (size_t)ET * 8;
    int*   mmax   = (int*)(denom + (size_t)n * 8);

    float* out = (float*)d_out;

    const int nodeGrid = (int)((nHC + 255) / 256);
    gat_init<<<nodeGrid, 256, 0, stream>>>(accum, denom, mmax, n);

    dim3 ggrid((n + 63) / 64, 3);
    gat_gemm3<<<ggrid, 256, 0, stream>>>(x, Wl, Wr, Wskip, xl, xr, skip, n);

    const int eblocks = 4096;
    gat_logits<<<eblocks, 256, 0, stream>>>(xl, xr, att, ei, E, n, logits, mmax);
    gat_accum<<<eblocks, 256, 0, stream>>>(xl, logits, mmax, ei, E, n, denom, accum);
    gat_finalize<<<nodeGrid, 256, 0, stream>>>(accum, denom, skip, bias, out, n);
}